// EmbedMeanField_64888365908123
// MI455X (gfx1250) — compile-verified
//
#include <hip/hip_runtime.h>

// ---------------------------------------------------------------------------
// EmbedMeanField on MI455X (gfx1250, wave32).
// Dense ops -> V_WMMA_F32_16X16X4_F32 (fp32, matches reference precision).
// Sparse ops -> coalesced per-lane GLOBAL_ATOMIC_ADD_F32 (resolves in L2;
// all pooled buffers are 12.8 MB << 192 MB L2).
// ---------------------------------------------------------------------------

typedef __attribute__((ext_vector_type(2))) float v2f;
typedef __attribute__((ext_vector_type(8))) float v8f;

static __device__ __forceinline__ v8f wmma_k4(v2f a, v2f b, v8f c) {
  // D = A(16x4,f32) * B(4x16,f32) + C(16x16,f32)
  return __builtin_amdgcn_wmma_f32_16x16x4_f32(
      /*neg_a=*/false, a, /*neg_b=*/false, b,
      /*c_mod=*/(short)0, c, /*reuse_a=*/false, /*reuse_b=*/false);
}

// VGPR layout assumptions (CDNA5 ISA 7.12.2, 32-bit data, wave32):
//   A 16x4 : lane L (row M=L%16) holds K = {2*(L/16), 2*(L/16)+1} in 2 VGPRs
//   B 4x16 : lane L (col N=L%16) VGPR r holds K = r + 2*(L/16)
//   C 16x16: lane L VGPR r holds (M = r + 8*(L/16), N = L%16)
// So for row-major X[m][k] (64-wide) the per-chunk A load is the float2 at
// v2f-index m*32 + 2*kc + hi, and for W[n][k] the B load is n*32 + 2*kc + hi.

__global__ void zero_f32(float4* __restrict__ p, int n4) {
  int i = blockIdx.x * blockDim.x + threadIdx.x;
  if (i < n4) p[i] = make_float4(0.f, 0.f, 0.f, 0.f);
}

// pool[dst[e]] += relu(edge_feat[e] @ w_e2l^T)   [E,16] x [16,64]
__global__ void edge_linear_scatter(const float* __restrict__ ef,
                                    const int* __restrict__ dst,
                                    const float* __restrict__ w,   // [64][16]
                                    float* __restrict__ pool,
                                    int nedges, int ntiles) {
  int wave = blockIdx.x * (blockDim.x >> 5) + (threadIdx.x >> 5);
  if (wave >= ntiles) return;                 // wave-uniform: EXEC stays all-1s
  int lane = threadIdx.x & 31;
  int hi = lane >> 4, lo = lane & 15;
  int e0 = wave * 16;

  int mA = e0 + lo; if (mA > nedges - 1) mA = nedges - 1;
  const v2f* ef2 = (const v2f*)ef;            // 16 floats/row = 8 v2f
  const v2f* w2  = (const v2f*)w;

  v2f a[4];
#pragma unroll
  for (int kc = 0; kc < 4; ++kc) a[kc] = ef2[mA * 8 + 2 * kc + hi];

  int drow[8];
#pragma unroll
  for (int r = 0; r < 8; ++r) {
    int e = e0 + r + 8 * hi; if (e > nedges - 1) e = nedges - 1;
    drow[r] = dst[e];
  }

  v8f c[4] = {};
#pragma unroll
  for (int t = 0; t < 4; ++t) {
    int n = 16 * t + lo;
#pragma unroll
    for (int kc = 0; kc < 4; ++kc)
      c[t] = wmma_k4(a[kc], w2[n * 8 + 2 * kc + hi], c[t]);
  }

#pragma unroll
  for (int t = 0; t < 4; ++t)
#pragma unroll
    for (int r = 0; r < 8; ++r) {
      int e = e0 + r + 8 * hi;
      if (e < nedges) {
        float v = c[t][r];
        v = v > 0.f ? v : 0.f;
        atomicAdd(&pool[(size_t)drow[r] * 64 + 16 * t + lo], v);
      }
    }
}

// stat = e2n @ w0^T ; cur = relu(stat)          [N,64] x [64,64]
__global__ void gemm_static(const float* __restrict__ e2n,
                            const float* __restrict__ w0,  // [64][64]
                            float* __restrict__ stat,
                            float* __restrict__ cur,
                            int nnodes, int ntiles) {
  int wave = blockIdx.x * (blockDim.x >> 5) + (threadIdx.x >> 5);
  if (wave >= ntiles) return;
  int lane = threadIdx.x & 31;
  int hi = lane >> 4, lo = lane & 15;
  int m0 = wave * 16;

  int mA = m0 + lo; if (mA > nnodes - 1) mA = nnodes - 1;
  const v2f* x2 = (const v2f*)e2n;
  const v2f* w2 = (const v2f*)w0;

  v2f a[16];
#pragma unroll
  for (int kc = 0; kc < 16; ++kc) a[kc] = x2[(size_t)mA * 32 + 2 * kc + hi];

  v8f c[4] = {};
#pragma unroll
  for (int t = 0; t < 4; ++t) {
    int n = 16 * t + lo;
#pragma unroll
    for (int kc = 0; kc < 16; ++kc)
      c[t] = wmma_k4(a[kc], w2[n * 32 + 2 * kc + hi], c[t]);
  }

#pragma unroll
  for (int t = 0; t < 4; ++t)
#pragma unroll
    for (int r = 0; r < 8; ++r) {
      int m = m0 + r + 8 * hi;
      if (m < nnodes) {
        float v = c[t][r];
        size_t o = (size_t)m * 64 + 16 * t + lo;
        stat[o] = v;
        cur[o]  = v > 0.f ? v : 0.f;
      }
    }
}

// n2n[dst[e]] += cur[src[e]]  (one lane per (edge,feature): fully coalesced
// 64-lane row reads + L2-resident non-returning float atomics)
__global__ void n2n_scatter(const float* __restrict__ cur,
                            const int* __restrict__ src,
                            const int* __restrict__ dst,
                            float* __restrict__ pool, int nedges) {
  int gid = blockIdx.x * blockDim.x + threadIdx.x;
  int e = gid >> 6;
  if (e >= nedges) return;
  int f = gid & 63;
  atomicAdd(&pool[(size_t)dst[e] * 64 + f], cur[(size_t)src[e] * 64 + f]);
}

// out = relu(stat + cur@w1^T + n2n@w2^T)
__global__ void gemm_update(const float* __restrict__ cur,
                            const float* __restrict__ n2n,
                            const float* __restrict__ stat,
                            const float* __restrict__ w1,
                            const float* __restrict__ w2,
                            float* __restrict__ out,
                            int nnodes, int ntiles) {
  int wave = blockIdx.x * (blockDim.x >> 5) + (threadIdx.x >> 5);
  if (wave >= ntiles) return;
  int lane = threadIdx.x & 31;
  int hi = lane >> 4, lo = lane & 15;
  int m0 = wave * 16;

  int mA = m0 + lo; if (mA > nnodes - 1) mA = nnodes - 1;
  const v2f* c2 = (const v2f*)cur;
  const v2f* p2 = (const v2f*)n2n;
  const v2f* w1v = (const v2f*)w1;
  const v2f* w2v = (const v2f*)w2;

  v2f a1[16], a2[16];
#pragma unroll
  for (int kc = 0; kc < 16; ++kc) {
    a1[kc] = c2[(size_t)mA * 32 + 2 * kc + hi];
    a2[kc] = p2[(size_t)mA * 32 + 2 * kc + hi];
  }

  // accumulate on top of static_message (bias-add folded into C init)
  v8f c[4];
#pragma unroll
  for (int t = 0; t < 4; ++t)
#pragma unroll
    for (int r = 0; r < 8; ++r) {
      int m = m0 + r + 8 * hi; if (m > nnodes - 1) m = nnodes - 1;
      c[t][r] = stat[(size_t)m * 64 + 16 * t + lo];
    }

#pragma unroll
  for (int t = 0; t < 4; ++t) {
    int n = 16 * t + lo;
#pragma unroll
    for (int kc = 0; kc < 16; ++kc) {
      c[t] = wmma_k4(a1[kc], w1v[n * 32 + 2 * kc + hi], c[t]);
      c[t] = wmma_k4(a2[kc], w2v[n * 32 + 2 * kc + hi], c[t]);
    }
  }

#pragma unroll
  for (int t = 0; t < 4; ++t)
#pragma unroll
    for (int r = 0; r < 8; ++r) {
      int m = m0 + r + 8 * hi;
      if (m < nnodes) {
        float v = c[t][r];
        out[(size_t)m * 64 + 16 * t + lo] = v > 0.f ? v : 0.f;
      }
    }
}

extern "C" void kernel_launch(void* const* d_in, const int* in_sizes, int n_in,
                              void* d_out, int out_size, void* d_ws, size_t ws_size,
                              hipStream_t stream) {
  const float* edge_feat = (const float*)d_in[0];
  const int*   src       = (const int*)d_in[1];
  const int*   dst       = (const int*)d_in[2];
  // d_in[3] = num_nodes (device scalar); derive N from out_size instead.
  const float* w_e2l     = (const float*)d_in[4];
  const float* w0        = (const float*)d_in[5];
  const float* w1        = (const float*)d_in[6];
  const float* w2        = (const float*)d_in[7];

  const int E = in_sizes[1];
  const int N = out_size / 64;
  float* out = (float*)d_out;

  // workspace: 4 buffers of N*64 f32 (~51 MB total)
  size_t nb = (size_t)N * 64;
  float* stat = (float*)d_ws;
  float* curA = stat + nb;
  float* curB = curA + nb;
  float* n2n  = curB + nb;
  float* e2n  = curB;              // dead before curB is first written

  const int B = 256;               // 8 waves/block
  int n4 = (int)(nb / 4);
  int zblk = (n4 + B - 1) / B;
  int etiles = (E + 15) / 16;
  int ntiles = (N + 15) / 16;
  int eblk = (etiles + 7) / 8;
  int nblk = (ntiles + 7) / 8;
  int sthreads = E * 64;
  int sblk = (sthreads + B - 1) / B;

  // level 0: edge linear -> e2n pool -> static message
  zero_f32<<<zblk, B, 0, stream>>>((float4*)e2n, n4);
  edge_linear_scatter<<<eblk, B, 0, stream>>>(edge_feat, dst, w_e2l, e2n, E, etiles);
  gemm_static<<<nblk, B, 0, stream>>>(e2n, w0, stat, curA, N, ntiles);

  // levels 1..3
  float* pin = curA;
  float* palt = curB;
  for (int lv = 0; lv < 3; ++lv) {
    zero_f32<<<zblk, B, 0, stream>>>((float4*)n2n, n4);
    n2n_scatter<<<sblk, B, 0, stream>>>(pin, src, dst, n2n, E);
    float* pout = (lv == 2) ? out : palt;
    gemm_update<<<nblk, B, 0, stream>>>(pin, n2n, stat, w1, w2, pout, N, ntiles);
    float* t = pin; pin = palt; palt = t;
  }
}